// UpdateUWithMLP_73469710565743
// MI455X (gfx1250) — compile-verified
//
#include <hip/hip_runtime.h>
#include <hip/hip_bf16.h>

#define HDIM   128
#define DTIME  0.1f
#define LNEPS  1e-5f

typedef float v2f __attribute__((ext_vector_type(2)));
typedef float v8f __attribute__((ext_vector_type(8)));

__device__ __forceinline__ float atomAddF(float* p, float v) {
#if defined(__AMDGCN__)
    return unsafeAtomicAdd(p, v);   // hardware global_atomic_add_f32
#else
    return atomicAdd(p, v);
#endif
}

// ---------------------------------------------------------------- zero fill
__global__ __launch_bounds__(256) void zero_f32(float* __restrict__ p, int n) {
    int i = blockIdx.x * 256 + threadIdx.x;
    if (i < n) p[i] = 0.0f;
}

// ------------------------------------------------- edge MLPs + segment sums
__global__ __launch_bounds__(256) void edge_mlp(
    const float* __restrict__ ea, const long long* __restrict__ ei, int E,
    const float* __restrict__ xw1, const float* __restrict__ xb1,
    const float* __restrict__ xw2, const float* __restrict__ xb2,
    const float* __restrict__ zw1, const float* __restrict__ zb1,
    const float* __restrict__ zw2, const float* __restrict__ zb2,
    float* __restrict__ dxinv, float* __restrict__ dzv,
    float* __restrict__ dsx, float* __restrict__ dsz)
{
    int e = blockIdx.x * 256 + threadIdx.x;
    if (e >= E) return;
    const float4* p = (const float4*)(ea + (size_t)e * 8);
    float4 f0 = p[0], f1 = p[1];
    float a[8] = { f0.x, f0.y, f0.z, f0.w, f1.x, f1.y, f1.z, f1.w };

    float vx = xb2[0], vz = zb2[0];
#pragma unroll
    for (int j = 0; j < 16; ++j) {
        float hx = xb1[j], hz = zb1[j];
#pragma unroll
        for (int i = 0; i < 8; ++i) {
            hx += a[i] * xw1[i * 16 + j];
            hz += a[i] * zw1[i * 16 + j];
        }
        vx += tanhf(hx) * xw2[j];
        vz += tanhf(hz) * zw2[j];
    }
    float dx = 1.0f / (vx + 1e-6f);   // vals for first spmm = 1/delta_x
    float dz = vz + 1e-6f;            // vals for second spmm = delta_z
    dxinv[e] = dx;
    dzv[e]   = dz;
    long long c = ei[(size_t)E + e];  // col
    atomAddF(&dsx[(int)c], dx);
    atomAddF(&dsz[(int)c], dz);
}

// -------------------------------------------- WMMA fp32 GEMM: C = A(NxK) * B(KxK)
// K = HDIM = 128. Block: 128 threads (4 waves), covers 16 rows x 64 cols.
// grid = ( ceil(N/16), 2 )
template <bool SQUARE>
__global__ __launch_bounds__(128) void gemm_wmma(
    const float* __restrict__ Amat, const float* __restrict__ Bmat,
    float* __restrict__ Cmat, int Nrows)
{
    __shared__ float Ws[128][68];   // 64-wide B panel, +4 pad
    __shared__ float Us[16][132];   // 16x128 A strip,  +4 pad (kills bank conflicts)

    const int tid      = threadIdx.x;
    const int colblock = blockIdx.y * 64;
    const int rowbase  = blockIdx.x * 16;

    // stage B panel (128x64)
    for (int s = tid; s < 2048; s += 128) {
        int k = s >> 4;
        int c = (s & 15) << 2;
        float4 v = *(const float4*)(Bmat + (size_t)k * HDIM + colblock + c);
        Ws[k][c] = v.x; Ws[k][c + 1] = v.y; Ws[k][c + 2] = v.z; Ws[k][c + 3] = v.w;
    }
    // stage A strip (16x128)
    for (int s = tid; s < 512; s += 128) {
        int r  = s >> 5;
        int c  = (s & 31) << 2;
        int gr = rowbase + r; if (gr >= Nrows) gr = Nrows - 1;
        float4 v = *(const float4*)(Amat + (size_t)gr * HDIM + c);
        Us[r][c] = v.x; Us[r][c + 1] = v.y; Us[r][c + 2] = v.z; Us[r][c + 3] = v.w;
    }
    __syncthreads();

    const int wave = tid >> 5;
    const int lane = tid & 31;
    const int half = lane >> 4;     // K half-select (ISA 16x4 A layout)
    const int l15  = lane & 15;     // M for A-frag, N for B-frag
    const int lcol = wave << 4;     // local col base in Ws

    v8f acc = {};
#pragma unroll
    for (int k0 = 0; k0 < HDIM; k0 += 4) {
        const int k = k0 + half * 2;
        v2f av, bv;
        av.x = Us[l15][k];            // A[M=l15][k], A[M=l15][k+1]
        av.y = Us[l15][k + 1];
        bv.x = Ws[k][lcol + l15];     // B[k][N=l15], B[k+1][N=l15]
        bv.y = Ws[k + 1][lcol + l15];
        acc = __builtin_amdgcn_wmma_f32_16x16x4_f32(
                  false, av, false, bv, (short)0, acc, false, false);
    }

    const int colbase = colblock + lcol;
#pragma unroll
    for (int j = 0; j < 8; ++j) {          // VGPR j -> rows {j, j+8} per half
        int m = rowbase + half * 8 + j;
        if (m < Nrows) {
            float v = acc[j];
            if (SQUARE) v = v * v;
            Cmat[(size_t)m * HDIM + colbase + l15] = v;
        }
    }
}

// ---------------------------------------- diag term: acc = fix(dsum) * M
__global__ __launch_bounds__(256) void init_acc(
    const float* __restrict__ uw2, const float* __restrict__ u,
    const float* __restrict__ dsx, const float* __restrict__ dsz,
    float* __restrict__ accx, float* __restrict__ accz, int n)
{
    int i = blockIdx.x * 256 + threadIdx.x;
    if (i >= n) return;
    int node = i >> 7;
    float dx = dsx[node]; dx += (dx == 0.0f) ? 1.0f : 0.0f;
    float dz = dsz[node]; dz += (dz == 0.0f) ? 1.0f : 0.0f;
    accx[i] = dx * uw2[i];
    accz[i] = dz * u[i];
}

// ------------------------- fused off-diagonal scatter for both spmms
// one wave per edge: gather uw2[row]+u[row] (512B each), scatter-add to acc[col]
__global__ __launch_bounds__(256) void spmm_edges(
    const long long* __restrict__ ei, int E,
    const float* __restrict__ dxinv, const float* __restrict__ dzv,
    const float* __restrict__ uw2, const float* __restrict__ u,
    float* __restrict__ accx, float* __restrict__ accz)
{
    int e = (blockIdx.x << 3) + (threadIdx.x >> 5);
    if (e >= E) return;
    int lane = threadIdx.x & 31;
    long long r = ei[e];
    long long c = ei[(size_t)E + e];
    float vx = dxinv[e], vz = dzv[e];
    const float* sx = uw2 + (size_t)r * HDIM;
    const float* sz = u   + (size_t)r * HDIM;
    float* dxp = accx + (size_t)c * HDIM;
    float* dzp = accz + (size_t)c * HDIM;
#pragma unroll
    for (int q = 0; q < 4; ++q) {
        int h = lane + (q << 5);
        atomAddF(&dxp[h], vx * sx[h]);
        atomAddF(&dzp[h], vz * sz[h]);
    }
}

// ------------------------------------- LN + silu + combine + clip (1 wave/row)
__device__ __forceinline__ float wred32(float v) {
#pragma unroll
    for (int m = 16; m; m >>= 1) v += __shfl_xor(v, m, 32);
    return v;
}
__device__ __forceinline__ float silu_f(float x) { return x / (1.0f + __expf(-x)); }

__global__ __launch_bounds__(256) void final_fuse(
    const float* __restrict__ u, const float* __restrict__ accx,
    const float* __restrict__ azt,
    const float* __restrict__ g1, const float* __restrict__ b1,
    const float* __restrict__ g2, const float* __restrict__ b2,
    const float* __restrict__ dgp, float* __restrict__ out, int N)
{
    int node = (blockIdx.x << 3) + (threadIdx.x >> 5);
    if (node >= N) return;
    int lane = threadIdx.x & 31;
    float dg = dgp[0];

    float xa[4], xz[4];
    float sa = 0.f, sa2 = 0.f, sz = 0.f, sz2 = 0.f;
#pragma unroll
    for (int q = 0; q < 4; ++q) {
        int h = lane + (q << 5);
        xa[q] = accx[(size_t)node * HDIM + h];
        xz[q] = azt [(size_t)node * HDIM + h];
        sa += xa[q]; sa2 += xa[q] * xa[q];
        sz += xz[q]; sz2 += xz[q] * xz[q];
    }
    sa = wred32(sa); sa2 = wred32(sa2);
    sz = wred32(sz); sz2 = wred32(sz2);
    const float inv = 1.0f / 128.0f;
    float mua = sa * inv, muz = sz * inv;
    float ra = rsqrtf(sa2 * inv - mua * mua + LNEPS);
    float rz = rsqrtf(sz2 * inv - muz * muz + LNEPS);
#pragma unroll
    for (int q = 0; q < 4; ++q) {
        int h = lane + (q << 5);
        float ya = silu_f((xa[q] - mua) * ra * g1[h] + b1[h]);
        float yz = silu_f((xz[q] - muz) * rz * g2[h] + b2[h]) + 1e-6f;
        float un = u[(size_t)node * HDIM + h] - DTIME * (ya + dg * yz);
        un = fminf(10.0f, fmaxf(-10.0f, un));
        out[(size_t)node * HDIM + h] = un;
    }
}

// ---------------------------------------------------------------- launcher
extern "C" void kernel_launch(void* const* d_in, const int* in_sizes, int n_in,
                              void* d_out, int out_size, void* d_ws, size_t ws_size,
                              hipStream_t stream)
{
    const float*     u    = (const float*)d_in[0];
    const long long* ei   = (const long long*)d_in[1];   // int64 [2,E]
    const float*     ea   = (const float*)d_in[2];
    const float*     xw1  = (const float*)d_in[4];
    const float*     xb1  = (const float*)d_in[5];
    const float*     xw2  = (const float*)d_in[6];
    const float*     xb2  = (const float*)d_in[7];
    const float*     zw1  = (const float*)d_in[8];
    const float*     zb1  = (const float*)d_in[9];
    const float*     zw2  = (const float*)d_in[10];
    const float*     zb2  = (const float*)d_in[11];
    const float*     Wm   = (const float*)d_in[12];
    const float*     AZw  = (const float*)d_in[13];
    const float*     n1g  = (const float*)d_in[14];
    const float*     n1b  = (const float*)d_in[15];
    const float*     n2g  = (const float*)d_in[16];
    const float*     n2b  = (const float*)d_in[17];
    const float*     dg   = (const float*)d_in[18];

    const int E = in_sizes[1] / 2;
    const int N = out_size / HDIM;
    const size_t NH = (size_t)N * HDIM;

    float* ws    = (float*)d_ws;
    float* uw2   = ws;                 // N*H, later reused for azt
    float* accx  = ws + NH;            // N*H
    float* accz  = ws + 2 * NH;        // N*H
    float* dxinv = ws + 3 * NH;        // E
    float* dzv   = dxinv + E;          // E
    float* dsx   = dzv + E;            // N  (dsz contiguous after)
    float* dsz   = dsx + N;            // N
    (void)dsz; (void)n_in; (void)ws_size;

    // 1. zero the segment sums (dsx & dsz are contiguous)
    zero_f32<<<(2 * N + 255) / 256, 256, 0, stream>>>(dsx, 2 * N);

    // 2. edge MLPs + segment sums
    edge_mlp<<<(E + 255) / 256, 256, 0, stream>>>(
        ea, ei, E, xw1, xb1, xw2, xb2, zw1, zb1, zw2, zb2,
        dxinv, dzv, dsx, dsz);

    // 3. uw2 = (u @ weight)^2   (WMMA fp32)
    dim3 ggrid((N + 15) / 16, 2);
    gemm_wmma<true><<<ggrid, 128, 0, stream>>>(u, Wm, uw2, N);

    // 4. diagonal terms
    init_acc<<<(int)((NH + 255) / 256), 256, 0, stream>>>(
        uw2, u, dsx, dsz, accx, accz, (int)NH);

    // 5. fused off-diagonal scatter for both spmms
    spmm_edges<<<(E + 7) / 8, 256, 0, stream>>>(
        ei, E, dxinv, dzv, uw2, u, accx, accz);

    // 6. azt = accz @ az_weight  (overwrites uw2, which is now dead)
    gemm_wmma<false><<<ggrid, 128, 0, stream>>>(accz, AZw, uw2, N);

    // 7. layernorms + silu + combine + clip
    final_fuse<<<(N + 7) / 8, 256, 0, stream>>>(
        u, accx, uw2, n1g, n1b, n2g, n2b, dg, (float*)d_out, N);
}